// QuantumKernelMethod_65481071409980
// MI455X (gfx1250) — compile-verified
//
#include <hip/hip_runtime.h>
#include <hip/hip_bf16.h>

// RBF kernel matrix on MI455X (gfx1250):
//   out[i][j] = exp(-gamma * max(|x_i|^2 + |y_j|^2 - 2 x_i.y_j, 0))
// n = m = 8192, d = 64, fp32.
//
// Roofline: output stores (256 MB) dominate -> ~11.5 us @ 23.3 TB/s.
// Inputs (4 MB) are L2-resident (192 MB L2). The 8.6 GFLOP GEMM is done
// with exact-precision V_WMMA_F32_16X16X4_F32 (A/B fp32), which is fast
// enough to stay under the store-bandwidth floor, so no precision is
// sacrificed.

typedef float v2f __attribute__((ext_vector_type(2)));
typedef float v8f __attribute__((ext_vector_type(8)));

#define TILE_D 64  // reference uses d = 64 (compile-time for full unroll)

// ---------------------------------------------------------------------------
// Pass 1: row squared-norms of x (N rows) and y (M rows) into workspace.
// 4 MB of reads total -> negligible. One thread per row, float4 loads.
// ---------------------------------------------------------------------------
__global__ void rbf_norms_kernel(const float* __restrict__ x,
                                 const float* __restrict__ y,
                                 float* __restrict__ xsq,
                                 float* __restrict__ ysq,
                                 int N, int M) {
    int t = blockIdx.x * blockDim.x + threadIdx.x;
    if (t >= N + M) return;
    const float* row = (t < N) ? (x + (size_t)t * TILE_D)
                               : (y + (size_t)(t - N) * TILE_D);
    float s = 0.0f;
#pragma unroll
    for (int i = 0; i < TILE_D / 4; ++i) {
        float4 v = ((const float4*)row)[i];
        s += v.x * v.x + v.y * v.y + v.z * v.z + v.w * v.w;
    }
    if (t < N) xsq[t] = s;
    else       ysq[t - N] = s;
}

// ---------------------------------------------------------------------------
// Pass 2: WMMA GEMM + fused RBF epilogue.
// Workgroup = 256 threads = 8 wave32 waves, arranged 4 (rows) x 2 (cols).
// Each wave computes a 32x32 output block = 2x2 subtiles of 16x16, using
// V_WMMA_F32_16X16X4_F32 over K = 64 (16 WMMA steps per subtile).
//
// f32 16x16x4 WMMA operand layout (ISA 7.12.2), per lane (m = lane%16,
// h = lane/16), K-step p (K base = 4p):
//   A: VGPR0 = x[r+m][4p+2h], VGPR1 = x[r+m][4p+2h+1]  -> one b64 load
//   B: VGPR0 = y[c+n][4p+2h], VGPR1 = y[c+n][4p+2h+1]  -> one b64 load
// C/D 16x16 f32 layout: lane L holds column n = L%16; VGPR v holds row
// v + 8*(L/16).
// ---------------------------------------------------------------------------
__global__ __launch_bounds__(256)
void rbf_wmma_kernel(const float* __restrict__ x,
                     const float* __restrict__ y,
                     const float* __restrict__ xsq,
                     const float* __restrict__ ysq,
                     const float* __restrict__ gptr,
                     float* __restrict__ out,
                     int N, int M) {
    const int lane = threadIdx.x & 31;
    const int wave = threadIdx.x >> 5;
    const int m    = lane & 15;   // row (A) / col (B) within 16
    const int h    = lane >> 4;   // K-half selector

    const int wr = wave & 3;      // wave row block (0..3)
    const int wc = wave >> 2;     // wave col block (0..1)

    const int r0 = blockIdx.x * 128 + wr * 32;  // 32 rows per wave
    const int c0 = blockIdx.y * 64  + wc * 32;  // 32 cols per wave

    // ---- load all K operands into registers (hoisted out of WMMA chain) ----
    v2f a[2][TILE_D / 4];  // two 16-row blocks
    v2f b[2][TILE_D / 4];  // two 16-col blocks
#pragma unroll
    for (int rb = 0; rb < 2; ++rb) {
        const float* xp = x + (size_t)(r0 + rb * 16 + m) * TILE_D + 2 * h;
#pragma unroll
        for (int p = 0; p < TILE_D / 4; ++p)
            a[rb][p] = *(const v2f*)(xp + 4 * p);
    }
#pragma unroll
    for (int cb = 0; cb < 2; ++cb) {
        const float* yp = y + (size_t)(c0 + cb * 16 + m) * TILE_D + 2 * h;
#pragma unroll
        for (int p = 0; p < TILE_D / 4; ++p)
            b[cb][p] = *(const v2f*)(yp + 4 * p);
    }

    // ---- 2x2 subtiles x 16 K-steps of exact f32 WMMA ----
    v8f acc[2][2] = {};
#pragma unroll
    for (int p = 0; p < TILE_D / 4; ++p) {
#pragma unroll
        for (int i = 0; i < 2; ++i) {
#pragma unroll
            for (int j = 0; j < 2; ++j) {
                acc[i][j] = __builtin_amdgcn_wmma_f32_16x16x4_f32(
                    /*neg_a=*/false, a[i][p],
                    /*neg_b=*/false, b[j][p],
                    /*c_mod=*/(short)0, acc[i][j],
                    /*reuse_a=*/false, /*reuse_b=*/false);
            }
        }
    }

    // ---- fused RBF epilogue: d = xsq + ysq - 2*xy; out = exp(-g*max(d,0)) --
    const float gamma = gptr[0];
    const int   n     = m;  // output column within subtile (C/D layout)

#pragma unroll
    for (int i = 0; i < 2; ++i) {
        const int rbase = r0 + 16 * i + 8 * h;  // rows rbase+0 .. rbase+7
        float xs[8];
#pragma unroll
        for (int v = 0; v < 8; ++v) xs[v] = xsq[rbase + v];
#pragma unroll
        for (int j = 0; j < 2; ++j) {
            const int col = c0 + 16 * j + n;
            const float ys = ysq[col];
            float* op = out + (size_t)rbase * M + col;
#pragma unroll
            for (int v = 0; v < 8; ++v) {
                float dist = xs[v] + ys - 2.0f * acc[i][j][v];
                dist = fmaxf(dist, 0.0f);
                op[(size_t)v * M] = __expf(-gamma * dist);
            }
        }
    }
}

// ---------------------------------------------------------------------------
extern "C" void kernel_launch(void* const* d_in, const int* in_sizes, int n_in,
                              void* d_out, int out_size, void* d_ws, size_t ws_size,
                              hipStream_t stream) {
    const float* x = (const float*)d_in[0];
    const float* y = (const float*)d_in[1];
    const float* g = (const float*)d_in[2];
    float* out = (float*)d_out;

    const int D = TILE_D;
    const int N = in_sizes[0] / D;  // 8192
    const int M = in_sizes[1] / D;  // 8192

    float* xsq = (float*)d_ws;
    float* ysq = xsq + N;

    // Pass 1: row norms (trivial)
    {
        int total = N + M;
        int blk = 256;
        rbf_norms_kernel<<<(total + blk - 1) / blk, blk, 0, stream>>>(
            x, y, xsq, ysq, N, M);
    }

    // Pass 2: WMMA GEMM + RBF epilogue.
    // WG tile = 128 rows x 64 cols (8 waves of 32x32).
    {
        dim3 grid(N / 128, M / 64);
        rbf_wmma_kernel<<<grid, 256, 0, stream>>>(
            x, y, xsq, ysq, g, out, N, M);
    }
}